// FastNGramLM_17282948399226
// MI455X (gfx1250) — compile-verified
//
#include <hip/hip_runtime.h>
#include <stdint.h>

#define VOCAB     1024
#define ARCS      16
#define THREADS   256
#define PER_TH    (VOCAB / THREADS)   // 4 vocab slots per thread
#define CHAIN     3                   // MAX_ORDER - 1 backoff hops

typedef float v4f __attribute__((ext_vector_type(4)));
typedef int   v4i __attribute__((ext_vector_type(4)));

#if defined(__gfx1250__) && __has_builtin(__builtin_amdgcn_global_load_async_to_lds_b32)
#define HAS_ASYNC_LDS 1
typedef __attribute__((address_space(1))) int* gint_p;
typedef __attribute__((address_space(3))) int* lint_p;
#else
#define HAS_ASYNC_LDS 0
#endif

__global__ __launch_bounds__(THREADS) void ngram_advance_kernel(
    const float* __restrict__ arcs_w,      // [V + (N-1)*A]
    const float* __restrict__ backoff_w,   // [N]
    const int*   __restrict__ ilabels,     // [V + (N-1)*A]
    const int*   __restrict__ to_states,   // [V + (N-1)*A]
    const int*   __restrict__ backoff_to,  // [N]
    const int*   __restrict__ st_start,    // [N]
    const int*   __restrict__ st_end,      // [N]
    const int*   __restrict__ states,      // [B]
    float*       __restrict__ out_scores,  // [B, V]
    int*         __restrict__ out_next)    // [B, V]
{
    __shared__ alignas(16) float s_score[VOCAB];
    __shared__ alignas(16) int   s_next[VOCAB];
    __shared__ int   s_lvState[CHAIN];
    __shared__ int   s_lvStart[CHAIN];
    __shared__ int   s_lvEnd[CHAIN];
    __shared__ float s_lvAcc[CHAIN];
    __shared__ float s_accF;

    const int b = blockIdx.x;
    const int t = threadIdx.x;

    // --- Stage dense start-state fallback tables into LDS (async DMA path) --
#if HAS_ASYNC_LDS
    #pragma unroll
    for (int k = 0; k < PER_TH; ++k) {
        const int v = t + k * THREADS;
        __builtin_amdgcn_global_load_async_to_lds_b32(
            (gint_p)(arcs_w + v),
            (lint_p)(&s_score[v]), 0, 0);
        __builtin_amdgcn_global_load_async_to_lds_b32(
            (gint_p)(to_states + v),
            (lint_p)(&s_next[v]), 0, 0);
    }
#else
    #pragma unroll
    for (int k = 0; k < PER_TH; ++k) {
        const int v = t + k * THREADS;
        s_score[v] = arcs_w[v];
        s_next[v]  = to_states[v];
    }
#endif

    // --- Thread 0 chases the backoff chain (overlaps the async copies) -----
    if (t == 0) {
        int   cur = states[b];
        float acc = 0.0f;
        #pragma unroll
        for (int lv = 0; lv < CHAIN; ++lv) {
            if (cur != 0) {                 // START == 0
                s_lvState[lv] = cur;
                s_lvStart[lv] = st_start[cur];
                s_lvEnd[lv]   = st_end[cur];
                s_lvAcc[lv]   = acc;        // acc BEFORE adding this state's backoff
                acc += backoff_w[cur];
                cur  = backoff_to[cur];
            } else {
                s_lvState[lv] = 0;          // level inactive
            }
        }
        s_accF = acc;                       // total backoff down to START
    }

#if HAS_ASYNC_LDS
  #if __has_builtin(__builtin_amdgcn_s_wait_asynccnt)
    __builtin_amdgcn_s_wait_asynccnt(0);
  #else
    asm volatile("s_wait_asynccnt 0" ::: "memory");
  #endif
#endif
    __syncthreads();

    // --- Wave 0 (lanes 0..15) gathers all sparse override arcs up front ----
    // Their VMEM latency overlaps the bias pass below.
    int   labv[CHAIN];
    float scv[CHAIN];
    int   nsv[CHAIN];
    bool  actv[CHAIN];
    if (t < ARCS) {
        #pragma unroll
        for (int lv = 0; lv < CHAIN; ++lv) {
            actv[lv] = false;
            const int st = s_lvState[lv];
            if (st != 0) {
                const int idx = s_lvStart[lv] + t;
                if (idx < s_lvEnd[lv]) {
                    actv[lv] = true;
                    labv[lv] = ilabels[idx];          // unique within a state
                    scv[lv]  = s_lvAcc[lv] + arcs_w[idx];
                    nsv[lv]  = to_states[idx];
                }
            }
        }
    }

    // --- Bias the dense fallback by the accumulated backoff weight (b128) --
    const float accF = s_accF;
    {
        v4f x = *(v4f*)(&s_score[4 * t]);
        x.x += accF; x.y += accF; x.z += accF; x.w += accF;
        *(v4f*)(&s_score[4 * t]) = x;
    }
    __syncthreads();

    // --- Sparse overrides, deepest backoff first (higher order overwrites).
    // All writers are lanes 0..15 of wave 0: intra-wave LDS ops are in-order,
    // so no barriers are needed between levels.
    if (t < ARCS) {
        #pragma unroll
        for (int lv = CHAIN - 1; lv >= 0; --lv) {
            if (actv[lv]) {
                s_score[labv[lv]] = scv[lv];
                s_next[labv[lv]]  = nsv[lv];
            }
        }
    }
    __syncthreads();

    // --- Coalesced, non-temporal b128 row write ------------------------------
    const size_t base = (size_t)b * VOCAB + 4 * t;
    {
        v4f xs = *(const v4f*)(&s_score[4 * t]);
        v4i xn = *(const v4i*)(&s_next[4 * t]);
        __builtin_nontemporal_store(xs, (v4f*)(&out_scores[base]));
        __builtin_nontemporal_store(xn, (v4i*)(&out_next[base]));
    }
}

extern "C" void kernel_launch(void* const* d_in, const int* in_sizes, int n_in,
                              void* d_out, int out_size, void* d_ws, size_t ws_size,
                              hipStream_t stream) {
    (void)n_in; (void)out_size; (void)d_ws; (void)ws_size;
    const float* arcs_w    = (const float*)d_in[0];
    const float* backoff_w = (const float*)d_in[1];
    const int*   ilabels   = (const int*)  d_in[2];
    const int*   to_states = (const int*)  d_in[3];
    const int*   backoff_to= (const int*)  d_in[4];
    const int*   st_start  = (const int*)  d_in[5];
    const int*   st_end    = (const int*)  d_in[6];
    const int*   states    = (const int*)  d_in[7];
    const int    B         = in_sizes[7];

    float* out_scores = (float*)d_out;
    int*   out_next   = (int*)((float*)d_out + (size_t)B * VOCAB);

    ngram_advance_kernel<<<B, THREADS, 0, stream>>>(
        arcs_w, backoff_w, ilabels, to_states, backoff_to,
        st_start, st_end, states, out_scores, out_next);
}